// DynamicGraphCore_51805895524418
// MI455X (gfx1250) — compile-verified
//
#include <hip/hip_runtime.h>
#include <math.h>

// Problem constants (reference shapes)
#define NB     4
#define NNODE  2048
#define SEQ    8192          // NB*NNODE
#define FH     64            // H == F == 64
#define G3H    192           // 3*H
#define KNN    8
#define EPB    9             // 8 knn edges + 1 self loop per node
#define OUTD   32
#define BN_EPS 1e-5f

typedef __attribute__((ext_vector_type(2))) float v2f;
typedef __attribute__((ext_vector_type(8))) float v8f;

// ---------------------------------------------------------------------------
// Fragment loader for V_WMMA_F32_16X16X4_F32.
// A (16x4, MxK):  lane l holds M=l%16, VGPR0 -> K=k0+2*(l/16), VGPR1 -> K+1
// B (4x16, KxN):  lane l holds N=l%16, VGPR0 -> K=k0+2*(l/16), VGPR1 -> K+1
// Both patterns read row (i0 + l%16) of a row-major [*, ld] matrix at
// consecutive k -> single float2 load.
// ---------------------------------------------------------------------------
static __device__ __forceinline__ v2f frag64(const float* __restrict__ m,
                                             int i0, int lane, int k0, int ld) {
  int i = i0 + (lane & 15);
  int k = k0 + ((lane >> 4) << 1);
  const float* p = m + (size_t)i * ld + k;
  v2f r;
  r.x = p[0];
  r.y = p[1];
  return r;
}

// ---------------------------------------------------------------------------
// Generic C = A(MxK=64) * B(NxK=64)^T (+bias), fp32 WMMA 16x16x4, K=64.
// grid = (M/16, N/16), block = 32 (one wave).
// ---------------------------------------------------------------------------
__global__ void gemm_nt64_wmma(const float* __restrict__ A, int lda,
                               const float* __restrict__ B,
                               const float* __restrict__ bias,
                               float* __restrict__ C, int ldc) {
  int lane = threadIdx.x;
  int row0 = blockIdx.x * 16;
  int col0 = blockIdx.y * 16;
  v8f c = {};
#pragma unroll
  for (int k0 = 0; k0 < FH; k0 += 4) {
    v2f a = frag64(A, row0, lane, k0, lda);
    v2f b = frag64(B, col0, lane, k0, FH);
    c = __builtin_amdgcn_wmma_f32_16x16x4_f32(false, a, false, b, (short)0, c,
                                              false, false);
  }
  float bv = bias ? bias[col0 + (lane & 15)] : 0.0f;
  int n = col0 + (lane & 15);
  int mbase = row0 + ((lane >> 4) << 3);
#pragma unroll
  for (int v = 0; v < 8; ++v) {
    C[(size_t)(mbase + v) * ldc + n] = c[v] + bv;
  }
}

// ---------------------------------------------------------------------------
// seq[s=b*2048+n][f] = x_seq[b, 7, f, n]   (only last "l" slice used)
// ---------------------------------------------------------------------------
__global__ void gather_seq(const float* __restrict__ x_seq,
                           float* __restrict__ seq) {
  int i = blockIdx.x * 256 + threadIdx.x;
  if (i >= SEQ * FH) return;
  int f = i & 63;
  int s = i >> 6;
  int b = s >> 11;
  int n = s & 2047;
  seq[i] = x_seq[(((size_t)b * 8 + 7) * FH + f) * NNODE + n];
}

// ---------------------------------------------------------------------------
// Sequential GRU scan over S=8192 (torch gate order r,z,n). Two blocks, one
// per GRU; 192 threads; Whh kept in LDS (stride 65 -> conflict-free).
// ---------------------------------------------------------------------------
static __device__ __forceinline__ float sigmoidf_(float x) {
  return 1.0f / (1.0f + expf(-x));
}

__global__ void __launch_bounds__(192)
gru_scan(const float* __restrict__ xp_s, const float* __restrict__ Whh_s,
         const float* __restrict__ bhh_s, float* __restrict__ out_s,
         const float* __restrict__ xp, const float* __restrict__ Whh,
         const float* __restrict__ bhh, float* __restrict__ out) {
  const float* XP  = (blockIdx.x == 0) ? xp_s  : xp;
  const float* WHH = (blockIdx.x == 0) ? Whh_s : Whh;
  const float* BHH = (blockIdx.x == 0) ? bhh_s : bhh;
  float*       OUT = (blockIdx.x == 0) ? out_s : out;

  __shared__ float sW[G3H * 65];
  __shared__ float sb[G3H];
  __shared__ float sh[FH];
  __shared__ float shp[G3H];
  __shared__ float sxt[G3H];

  int t = threadIdx.x;  // 0..191
  for (int i = t; i < G3H * FH; i += G3H) {
    int r = i / FH, k = i - r * FH;
    sW[r * 65 + k] = WHH[i];
  }
  sb[t] = BHH[t];
  if (t < FH) sh[t] = 0.0f;
  __syncthreads();

  const float* wrow = &sW[t * 65];
  for (int s = 0; s < SEQ; ++s) {
    float acc = sb[t];
#pragma unroll 8
    for (int k = 0; k < FH; ++k) acc += wrow[k] * sh[k];
    shp[t] = acc;
    sxt[t] = XP[(size_t)s * G3H + t];
    __syncthreads();
    if (t < FH) {
      float r  = sigmoidf_(sxt[t] + shp[t]);
      float z  = sigmoidf_(sxt[FH + t] + shp[FH + t]);
      float ng = tanhf(sxt[2 * FH + t] + r * shp[2 * FH + t]);
      float hn = (1.0f - z) * ng + z * sh[t];
      sh[t] = hn;
      OUT[(size_t)s * FH + t] = hn;
    }
    __syncthreads();
  }
}

// ---------------------------------------------------------------------------
// Fused Gram matrix (sim = h h^T) + masked diagonal + streaming top-8.
// grid = (2048/16 row tiles, 4 batches), block = 32 (one wave).
// A-frags cached in registers across all 128 column tiles.
// ---------------------------------------------------------------------------
__global__ void __launch_bounds__(32)
sim_topk(const float* __restrict__ h, float* __restrict__ vals_out,
         int* __restrict__ idx_out) {
  int lane = threadIdx.x;
  int row0 = blockIdx.x * 16;
  int b    = blockIdx.y;
  const float* hb = h + (size_t)b * NNODE * FH;

  v2f a[16];
#pragma unroll
  for (int ks = 0; ks < 16; ++ks) a[ks] = frag64(hb, row0, lane, ks * 4, FH);

  float tv[KNN];
  int   ti[KNN];
#pragma unroll
  for (int j = 0; j < KNN; ++j) { tv[j] = -3.0e38f; ti[j] = 0; }

  __shared__ float smem[16][17];

  for (int ct = 0; ct < NNODE / 16; ++ct) {
    int col0 = ct * 16;
    v8f c = {};
#pragma unroll
    for (int ks = 0; ks < 16; ++ks) {
      v2f bf = frag64(hb, col0, lane, ks * 4, FH);
      c = __builtin_amdgcn_wmma_f32_16x16x4_f32(false, a[ks], false, bf,
                                                (short)0, c, false, false);
    }
    // stage tile to LDS with the diagonal masked to -1e9
    int n = lane & 15;
#pragma unroll
    for (int v = 0; v < 8; ++v) {
      int m = v + ((lane >> 4) << 3);
      float val = c[v];
      if (row0 + m == col0 + n) val = -1.0e9f;
      smem[m][n] = val;
    }
    __syncthreads();
    if (lane < 16) {  // lane owns row (row0+lane); serial top-8 insertion
#pragma unroll
      for (int nn = 0; nn < 16; ++nn) {
        float val = smem[lane][nn];
        if (val > tv[KNN - 1]) {
          tv[KNN - 1] = val;
          ti[KNN - 1] = col0 + nn;
#pragma unroll
          for (int j = KNN - 1; j > 0; --j) {
            if (tv[j] > tv[j - 1]) {
              float tf = tv[j]; tv[j] = tv[j - 1]; tv[j - 1] = tf;
              int   tx = ti[j]; ti[j] = ti[j - 1]; ti[j - 1] = tx;
            }
          }
        }
      }
    }
    __syncthreads();
  }

  if (lane < 16) {
    size_t base = ((size_t)b * NNODE + row0 + lane) * KNN;
#pragma unroll
    for (int j = 0; j < KNN; ++j) {
      vals_out[base + j] = tv[j];
      idx_out[base + j]  = ti[j];
    }
  }
}

// ---------------------------------------------------------------------------
// Graph kernels
// ---------------------------------------------------------------------------
__global__ void zero_f32(float* __restrict__ p, int n) {
  int i = blockIdx.x * 256 + threadIdx.x;
  if (i < n) p[i] = 0.0f;
}

__global__ void compute_deg(const float* __restrict__ vals,
                            const int* __restrict__ idx,
                            float* __restrict__ deg) {
  int e = blockIdx.x * 256 + threadIdx.x;
  if (e >= NB * NNODE * EPB) return;
  int b    = e / (NNODE * EPB);
  int rem  = e - b * (NNODE * EPB);
  int r    = rem / EPB;
  int slot = rem - r * EPB;
  int c; float w;
  if (slot < KNN) {
    size_t o = ((size_t)b * NNODE + r) * KNN + slot;
    c = idx[o]; w = vals[o];
  } else {
    c = r; w = 1.0f;
  }
  atomicAdd(&deg[b * NNODE + c], w);
}

__global__ void gcn_scatter(const float* __restrict__ vals,
                            const int* __restrict__ idx,
                            const float* __restrict__ deg,
                            const float* __restrict__ xw,
                            float* __restrict__ acc) {
  long long i = (long long)blockIdx.x * 256 + threadIdx.x;
  if (i >= (long long)NB * NNODE * EPB * FH) return;
  int f = (int)(i & 63);
  int e = (int)(i >> 6);
  int b    = e / (NNODE * EPB);
  int rem  = e - b * (NNODE * EPB);
  int r    = rem / EPB;
  int slot = rem - r * EPB;
  int c; float w;
  if (slot < KNN) {
    size_t o = ((size_t)b * NNODE + r) * KNN + slot;
    c = idx[o]; w = vals[o];
  } else {
    c = r; w = 1.0f;
  }
  float dr = deg[b * NNODE + r];
  float dc = deg[b * NNODE + c];
  float ir = rsqrtf(dr); if (isinf(ir)) ir = 0.0f;
  float ic = rsqrtf(dc); if (isinf(ic)) ic = 0.0f;
  float norm = ir * w * ic;
  float contrib = norm * xw[((size_t)b * NNODE + r) * FH + f];
  atomicAdd(&acc[((size_t)b * NNODE + c) * FH + f], contrib);
}

__global__ void bias_relu(const float* __restrict__ acc,
                          const float* __restrict__ bias,
                          float* __restrict__ dst, int n) {
  int i = blockIdx.x * 256 + threadIdx.x;
  if (i >= n) return;
  float v = acc[i] + bias[i & 63];
  dst[i] = v > 0.0f ? v : 0.0f;
}

// ---------------------------------------------------------------------------
// BatchNorm (training-mode biased stats over 8192 rows), folded to scale/shift
// ---------------------------------------------------------------------------
__global__ void bn_stats(const float* __restrict__ flat,
                         const float* __restrict__ gamma,
                         const float* __restrict__ beta,
                         float* __restrict__ scale,
                         float* __restrict__ shift) {
  int f = blockIdx.x;   // feature 0..63
  int t = threadIdx.x;  // 0..255
  float s = 0.0f, sq = 0.0f;
  for (int r = t; r < SEQ; r += 256) {
    float v = flat[(size_t)r * FH + f];
    s += v;
    sq += v * v;
  }
  __shared__ float ss[256], ssq[256];
  ss[t] = s; ssq[t] = sq;
  __syncthreads();
  for (int o = 128; o > 0; o >>= 1) {
    if (t < o) { ss[t] += ss[t + o]; ssq[t] += ssq[t + o]; }
    __syncthreads();
  }
  if (t == 0) {
    float mu  = ss[0] * (1.0f / SEQ);
    float var = ssq[0] * (1.0f / SEQ) - mu * mu;
    float sc  = gamma[f] / sqrtf(var + BN_EPS);
    scale[f] = sc;
    shift[f] = beta[f] - mu * sc;
  }
}

__global__ void bn_apply(const float* __restrict__ flat,
                         const float* __restrict__ scale,
                         const float* __restrict__ shift,
                         float* __restrict__ normed, int n) {
  int i = blockIdx.x * 256 + threadIdx.x;
  if (i >= n) return;
  int f = i & 63;
  normed[i] = flat[i] * scale[f] + shift[f];
}

// ---------------------------------------------------------------------------
// Orchestration
// ---------------------------------------------------------------------------
extern "C" void kernel_launch(void* const* d_in, const int* in_sizes, int n_in,
                              void* d_out, int out_size, void* d_ws,
                              size_t ws_size, hipStream_t stream) {
  const float* x_seq = (const float*)d_in[0];
  const float* Wih_s = (const float*)d_in[1];
  const float* Whh_s = (const float*)d_in[2];
  const float* bih_s = (const float*)d_in[3];
  const float* bhh_s = (const float*)d_in[4];
  const float* Wih   = (const float*)d_in[5];
  const float* Whh   = (const float*)d_in[6];
  const float* bih   = (const float*)d_in[7];
  const float* bhh   = (const float*)d_in[8];
  const float* W1    = (const float*)d_in[9];
  const float* b1    = (const float*)d_in[10];
  const float* W2    = (const float*)d_in[11];
  const float* b2    = (const float*)d_in[12];
  const float* gamma = (const float*)d_in[13];
  const float* beta  = (const float*)d_in[14];
  const float* Wp    = (const float*)d_in[15];
  const float* bp    = (const float*)d_in[16];

  // Workspace layout (floats); total ~7.5M floats (~30 MB)
  float* ws     = (float*)d_ws;
  float* seq    = ws;                       // 524288
  float* xp_s   = seq + SEQ * FH;           // 1572864
  float* xp     = xp_s + SEQ * G3H;         // 1572864
  float* h_s    = xp + SEQ * G3H;           // 524288
  float* h_out  = h_s + SEQ * FH;           // 524288
  float* vals   = h_out + SEQ * FH;         // 65536
  int*   idx    = (int*)(vals + SEQ * KNN); // 65536
  float* deg    = (float*)(idx + SEQ * KNN);// 8192
  float* xw     = deg + SEQ;                // 524288
  float* acc    = xw + SEQ * FH;            // 524288
  float* o1     = acc + SEQ * FH;           // 524288
  float* o2     = o1 + SEQ * FH;            // 524288
  float* scale  = o2 + SEQ * FH;            // 64
  float* shift  = scale + FH;               // 64
  float* normed = shift + FH;               // 524288

  // 1) gather last time-slice, transpose to [8192, 64]
  gather_seq<<<(SEQ * FH + 255) / 256, 256, 0, stream>>>(x_seq, seq);

  // 2) GRU input projections (WMMA): xp = seq @ Wih^T + bih   [8192,192]
  gemm_nt64_wmma<<<dim3(SEQ / 16, G3H / 16), 32, 0, stream>>>(
      seq, FH, Wih_s, bih_s, xp_s, G3H);
  gemm_nt64_wmma<<<dim3(SEQ / 16, G3H / 16), 32, 0, stream>>>(
      seq, FH, Wih, bih, xp, G3H);

  // 3) sequential GRU recurrences (both GRUs concurrently, 1 block each)
  gru_scan<<<2, 192, 0, stream>>>(xp_s, Whh_s, bhh_s, h_s, xp, Whh, bhh, h_out);

  // 4) fused Gram matrix + top-8 (WMMA, sim never materialized)
  sim_topk<<<dim3(NNODE / 16, NB), 32, 0, stream>>>(h_s, vals, idx);

  // 5) degrees
  zero_f32<<<(SEQ + 255) / 256, 256, 0, stream>>>(deg, SEQ);
  compute_deg<<<(NB * NNODE * EPB + 255) / 256, 256, 0, stream>>>(vals, idx, deg);

  // 6) GCN layer 1: xw = h_out @ W1^T (WMMA), scatter, bias+relu
  gemm_nt64_wmma<<<dim3(SEQ / 16, FH / 16), 32, 0, stream>>>(
      h_out, FH, W1, nullptr, xw, FH);
  zero_f32<<<(SEQ * FH + 255) / 256, 256, 0, stream>>>(acc, SEQ * FH);
  gcn_scatter<<<(NB * NNODE * EPB * FH + 255) / 256, 256, 0, stream>>>(
      vals, idx, deg, xw, acc);
  bias_relu<<<(SEQ * FH + 255) / 256, 256, 0, stream>>>(acc, b1, o1, SEQ * FH);

  // 7) GCN layer 2
  gemm_nt64_wmma<<<dim3(SEQ / 16, FH / 16), 32, 0, stream>>>(
      o1, FH, W2, nullptr, xw, FH);
  zero_f32<<<(SEQ * FH + 255) / 256, 256, 0, stream>>>(acc, SEQ * FH);
  gcn_scatter<<<(NB * NNODE * EPB * FH + 255) / 256, 256, 0, stream>>>(
      vals, idx, deg, xw, acc);
  bias_relu<<<(SEQ * FH + 255) / 256, 256, 0, stream>>>(acc, b2, o2, SEQ * FH);

  // 8) BatchNorm folded to scale/shift, applied elementwise
  bn_stats<<<FH, 256, 0, stream>>>(o2, gamma, beta, scale, shift);
  bn_apply<<<(SEQ * FH + 255) / 256, 256, 0, stream>>>(o2, scale, shift,
                                                       normed, SEQ * FH);

  // 9) predictor (WMMA): d_out = normed @ Wp^T + bp    [8192, 32]
  gemm_nt64_wmma<<<dim3(SEQ / 16, OUTD / 16), 32, 0, stream>>>(
      normed, FH, Wp, bp, (float*)d_out, OUTD);
}